// BSplineNN_32796370272360
// MI455X (gfx1250) — compile-verified
//
#include <hip/hip_runtime.h>
#include <hip/hip_bf16.h>

typedef __attribute__((ext_vector_type(2))) float v2f;
typedef __attribute__((ext_vector_type(4))) float v4f;
typedef __attribute__((ext_vector_type(8))) float v8f;

// B-spline evaluation, exploiting cubic-spline locality: only 4 basis
// functions are nonzero at a scalar x, so we gather 4 coefficient rows
// instead of 64 (268MB -> ~22MB of HBM traffic).
// Data movement: GLOBAL_LOAD_ASYNC_TO_LDS_B128 stages the 4x256 coefficient
// window straight into LDS (ASYNCcnt-tracked, no VGPR round trip), overlapped
// with the de Boor basis computation.
// Contraction out[c] = sum_k basis[k]*win[k][c] done with
// V_WMMA_F32_16X16X4_F32 (A = basis replicated over 16 rows, B = 4x16
// channel tile, every D row = the 16-channel result).
__global__ __launch_bounds__(256) void bspline_wmma_kernel(
    const float* __restrict__ coef,   // [B, n, C]
    const float* __restrict__ knots,  // [B, T]
    const float* __restrict__ inpce,  // [B, 1]
    float* __restrict__ out,          // [B, C]
    int n, int C, int T)
{
    const int b   = blockIdx.x;
    const int tid = threadIdx.x;

    __shared__ int   s_j;
    __shared__ float s_basis[4];
    __shared__ __align__(16) float s_win[4][256];   // 4-row coef window, zero-filled

    if (tid == 0) s_j = -1;
    __syncthreads();

    const float  x  = inpce[b];
    const float* kt = knots + (size_t)b * T;

    // ---- Phase 1: lane-parallel knot-interval search (T-1 = 67 intervals).
    // Half-open intervals of sorted knots are disjoint -> at most one writer.
    if (tid < T - 1) {
        float t0 = kt[tid], t1 = kt[tid + 1];
        if (t0 <= x && x < t1) s_j = tid;
    }
    __syncthreads();
    const int j = s_j;   // -1 => x outside [t0, t_{T-1}) => output is zero

    // ---- Phase 2a: stage coef rows j-3..j into LDS with the async DMA path.
    // Each thread owns one float4 (16B) of the 4KB window. Valid rows use
    // GLOBAL_LOAD_ASYNC_TO_LDS_B128 (memory -> LDS directly); rows whose basis
    // index falls outside [0, n) are zero-filled with a normal LDS store.
    {
        int r  = tid >> 6;          // 0..3  : window row
        int c4 = (tid & 63) << 2;   // 0..252: channel group (float4)
        int i  = j - 3 + r;         // basis/coefficient row index
        // Generic shared pointer: low 32 bits are the LDS byte address
        // (ISA 10.2 aperture rule).
        unsigned lds_addr = (unsigned)(uintptr_t)(&s_win[r][c4]);
        if (j >= 0 && i >= 0 && i < n) {
            const float* gp = coef + ((size_t)b * n + i) * C + c4;
            asm volatile("global_load_async_to_lds_b128 %0, %1, off"
                         :: "v"(lds_addr), "v"(gp)
                         : "memory");
        } else {
            v4f z = {0.f, 0.f, 0.f, 0.f};
            *(v4f*)(&s_win[r][c4]) = z;
        }
    }

    // ---- Phase 2b: de Boor / Cox-de Boor on the 8-knot support window.
    // Identical math to the reference recurrence (incl. d==0 guards); knot
    // index clamping only affects entries that are discarded below.
    // Runs concurrently with the async LDS fill above.
    if (tid == 0) {
        float bas[4] = {0.f, 0.f, 0.f, 0.f};
        if (j >= 0) {
            float t[8];
#pragma unroll
            for (int r = 0; r < 8; ++r) {
                int idx = j - 3 + r;
                idx = idx < 0 ? 0 : (idx > T - 1 ? T - 1 : idx);
                t[r] = kt[idx];
            }
            // Level-0 seeds: local m <-> basis index i = j-3+m; only i==j is 1.
            float Nv[7] = {0.f, 0.f, 0.f, 1.f, 0.f, 0.f, 0.f};
#pragma unroll
            for (int kk = 1; kk <= 3; ++kk) {
#pragma unroll
                for (int m = 0; m + kk < 7; ++m) {
                    float d1 = t[m + kk]     - t[m];
                    float d2 = t[m + kk + 1] - t[m + 1];
                    float w1 = (d1 == 0.f) ? 0.f : (x - t[m]) / d1;
                    float w2 = (d2 == 0.f) ? 0.f : (t[m + kk + 1] - x) / d2;
                    Nv[m] = w1 * Nv[m] + w2 * Nv[m + 1];
                }
            }
#pragma unroll
            for (int r = 0; r < 4; ++r) {
                int i = j - 3 + r;
                bas[r] = (i >= 0 && i < n) ? Nv[r] : 0.f;
            }
        }
        s_basis[0] = bas[0]; s_basis[1] = bas[1];
        s_basis[2] = bas[2]; s_basis[3] = bas[3];
    }

    // Drain this wave's async LDS writes before the block barrier so the
    // window is visible to all waves (ASYNCcnt decrements on LDS completion).
    asm volatile("s_wait_asynccnt 0x0" ::: "memory");
    __syncthreads();

    // ---- Phase 3: WMMA contraction. Wave w covers channels [w*32, w*32+32).
    // A 16x4 f32 layout (ISA 7.12.2): lanes 0-15 hold K=0 (v0) and K=1 (v1),
    // lanes 16-31 hold K=2,K=3 -> replicate basis so every A-row == basis.
    const int wave = tid >> 5;
    const int lane = tid & 31;
    const int half = lane >> 4;   // 0 -> K={0,1}, 1 -> K={2,3}
    const int ncol = lane & 15;

    v2f a;
    a.x = s_basis[half * 2 + 0];
    a.y = s_basis[half * 2 + 1];

#pragma unroll
    for (int piece = 0; piece < 2; ++piece) {
        const int c0 = wave * 32 + piece * 16;
        // B 4x16: row K striped across lanes within a VGPR, mirroring A's
        // K split: lanes 0-15 carry K=0 (v0) / K=1 (v1), lanes 16-31 K=2/K=3.
        v2f bm;
        bm.x = s_win[half * 2 + 0][c0 + ncol];
        bm.y = s_win[half * 2 + 1][c0 + ncol];

        v8f acc = {};
        acc = __builtin_amdgcn_wmma_f32_16x16x4_f32(
            /*neg_a=*/false, a, /*neg_b=*/false, bm,
            /*c_mod=*/(short)0, acc, /*reuse_a=*/false, /*reuse_b=*/false);

        // Every D row is identical; VGPR 0 holds M=0 (lanes 0-15).
        if (lane < 16) out[(size_t)b * C + c0 + lane] = acc[0];
    }
}

extern "C" void kernel_launch(void* const* d_in, const int* in_sizes, int n_in,
                              void* d_out, int out_size, void* d_ws, size_t ws_size,
                              hipStream_t stream) {
    const float* coef  = (const float*)d_in[0];  // [B, n, C] fp32
    const float* knots = (const float*)d_in[1];  // [B, T]    fp32
    const float* inpce = (const float*)d_in[2];  // [B, 1]    fp32
    float*       outp  = (float*)d_out;          // [B, C]    fp32

    const int B = in_sizes[2];                   // inpce has B elements
    const int C = out_size / B;                  // 256
    const int n = in_sizes[0] / (B * C);         // 64
    const int T = in_sizes[1] / B;               // 68

    bspline_wmma_kernel<<<B, 256, 0, stream>>>(coef, knots, inpce, outp, n, C, T);
}